// DocumentGNN_39453569581540
// MI455X (gfx1250) — compile-verified
//
#include <hip/hip_runtime.h>
#include <hip/hip_bf16.h>
#include <math.h>

typedef float v2f __attribute__((ext_vector_type(2)));
typedef float v8f __attribute__((ext_vector_type(8)));

#define N_NODES  100000
#define N_EDGES  3200000
#define N_GRAPHS 256
#define NTILES   (N_NODES / 16)   // 6250, exact (no tail)

// ---------------------------------------------------------------------------
// Degree / normalization
// ---------------------------------------------------------------------------
__global__ void init_deg_kernel(float* __restrict__ deg) {
    int i = blockIdx.x * blockDim.x + threadIdx.x;
    if (i < N_NODES) deg[i] = 1.0f;             // self-loop contribution
}

__global__ void count_deg_kernel(const int* __restrict__ ei, float* __restrict__ deg) {
    int e = blockIdx.x * blockDim.x + threadIdx.x;
    if (e < N_EDGES) atomicAdd(&deg[ei[N_EDGES + e]], 1.0f);   // dst row
}

__global__ void dinv_kernel(const float* __restrict__ deg, float* __restrict__ dinv) {
    int i = blockIdx.x * blockDim.x + threadIdx.x;
    if (i < N_NODES) dinv[i] = rsqrtf(deg[i]);
}

// ---------------------------------------------------------------------------
// Layer 1: h1 = x @ W1   (8 -> 16), seed agg1 = h1 * dinv^2
// One wave per 16-node tile. WMMA f32 16x16x4, K=8 in two chunks.
// A layout (ISA 7.12.2, 32-bit A 16x4): lane<16 -> M=lane, K={k0,k0+1};
// lane>=16 -> M=lane-16, K={k0+2,k0+3}. B mirrors with N=lane&15.
// C: VGPR v -> (M = v + 8*(lane>>4), N = lane&15).
// ---------------------------------------------------------------------------
__global__ void transform1_kernel(const float* __restrict__ x,
                                  const float* __restrict__ W1,
                                  const float* __restrict__ dinv,
                                  float* __restrict__ h1,
                                  float* __restrict__ agg1) {
    int wave = threadIdx.x >> 5;
    int lane = threadIdx.x & 31;
    int tile = blockIdx.x * 8 + wave;
    if (tile >= NTILES) return;                 // wave-uniform guard

    int row  = lane & 15;
    int half = lane >> 4;
    int nb   = tile * 16;
    int node = nb + row;

    v8f c = {};
#pragma unroll
    for (int k0 = 0; k0 < 8; k0 += 4) {
        int kb = k0 + half * 2;
        v2f a, b;
        a.x = x[node * 8 + kb + 0];
        a.y = x[node * 8 + kb + 1];
        b.x = W1[(kb + 0) * 16 + row];          // W1 is [8,16] row-major
        b.y = W1[(kb + 1) * 16 + row];
        c = __builtin_amdgcn_wmma_f32_16x16x4_f32(false, a, false, b,
                                                  (short)0, c, false, false);
    }
#pragma unroll
    for (int v = 0; v < 8; ++v) {
        int   m   = nb + v + half * 8;
        float val = c[v];
        h1[m * 16 + row] = val;
        float di = dinv[m];
        agg1[m * 16 + row] = val * di * di;     // self-loop term
    }
}

// ---------------------------------------------------------------------------
// Layer 1 edge scatter: agg1[dst] += h1[src] * (dinv[src]*dinv[dst])
// Thread per (edge, quarter of 16 features): float4 gather + 4 atomics.
// ---------------------------------------------------------------------------
__global__ void scatter1_kernel(const int* __restrict__ ei,
                                const float* __restrict__ dinv,
                                const float* __restrict__ h1,
                                float* __restrict__ agg1) {
    int tid = blockIdx.x * blockDim.x + threadIdx.x;
    if (tid >= N_EDGES * 4) return;
    int e = tid >> 2;
    int q = tid & 3;
    int s = ei[e];
    int d = ei[N_EDGES + e];
    float coef = dinv[s] * dinv[d];
    float4 hv  = *(const float4*)(h1 + s * 16 + q * 4);
    float* out = agg1 + d * 16 + q * 4;
    atomicAdd(out + 0, hv.x * coef);
    atomicAdd(out + 1, hv.y * coef);
    atomicAdd(out + 2, hv.z * coef);
    atomicAdd(out + 3, hv.w * coef);
}

// ---------------------------------------------------------------------------
// Layer 2: h2 = relu(agg1 + b1) @ W2   (16 -> 32), seed agg2 = h2 * dinv^2
// Two 16-column output blocks, K=16 in four chunks -> 8 WMMAs per tile.
// ---------------------------------------------------------------------------
__global__ void transform2_kernel(const float* __restrict__ agg1,
                                  const float* __restrict__ b1,
                                  const float* __restrict__ W2,
                                  const float* __restrict__ dinv,
                                  float* __restrict__ h2,
                                  float* __restrict__ agg2) {
    int wave = threadIdx.x >> 5;
    int lane = threadIdx.x & 31;
    int tile = blockIdx.x * 8 + wave;
    if (tile >= NTILES) return;

    int row  = lane & 15;
    int half = lane >> 4;
    int nb   = tile * 16;
    int node = nb + row;

    // Load the four A chunks (relu(agg1 + b1)) once.
    v2f A[4];
#pragma unroll
    for (int kc = 0; kc < 4; ++kc) {
        int kb = kc * 4 + half * 2;
        A[kc].x = fmaxf(agg1[node * 16 + kb + 0] + b1[kb + 0], 0.0f);
        A[kc].y = fmaxf(agg1[node * 16 + kb + 1] + b1[kb + 1], 0.0f);
    }

    v8f c0 = {}, c1 = {};
#pragma unroll
    for (int kc = 0; kc < 4; ++kc) {
        int kb = kc * 4 + half * 2;
        v2f b;
        b.x = W2[(kb + 0) * 32 + row];          // cols 0..15, W2 is [16,32]
        b.y = W2[(kb + 1) * 32 + row];
        c0 = __builtin_amdgcn_wmma_f32_16x16x4_f32(false, A[kc], false, b,
                                                   (short)0, c0, false, false);
        b.x = W2[(kb + 0) * 32 + 16 + row];     // cols 16..31
        b.y = W2[(kb + 1) * 32 + 16 + row];
        c1 = __builtin_amdgcn_wmma_f32_16x16x4_f32(false, A[kc], false, b,
                                                   (short)0, c1, false, false);
    }
#pragma unroll
    for (int v = 0; v < 8; ++v) {
        int   m  = nb + v + half * 8;
        float di = dinv[m];
        float d2 = di * di;
        float v0 = c0[v];
        float v1 = c1[v];
        h2[m * 32 + row]        = v0;
        h2[m * 32 + 16 + row]   = v1;
        agg2[m * 32 + row]      = v0 * d2;
        agg2[m * 32 + 16 + row] = v1 * d2;
    }
}

// ---------------------------------------------------------------------------
// Layer 2 edge scatter: agg2[dst] += h2[src] * coef  (32 features, 8 quarters)
// ---------------------------------------------------------------------------
__global__ void scatter2_kernel(const int* __restrict__ ei,
                                const float* __restrict__ dinv,
                                const float* __restrict__ h2,
                                float* __restrict__ agg2) {
    int tid = blockIdx.x * blockDim.x + threadIdx.x;
    if (tid >= N_EDGES * 8) return;
    int e = tid >> 3;
    int q = tid & 7;
    int s = ei[e];
    int d = ei[N_EDGES + e];
    float coef = dinv[s] * dinv[d];
    float4 hv  = *(const float4*)(h2 + s * 32 + q * 4);
    float* out = agg2 + d * 32 + q * 4;
    atomicAdd(out + 0, hv.x * coef);
    atomicAdd(out + 1, hv.y * coef);
    atomicAdd(out + 2, hv.z * coef);
    atomicAdd(out + 3, hv.w * coef);
}

// ---------------------------------------------------------------------------
// Zero the pooled accumulators (d_ws is poisoned, not zeroed)
// ---------------------------------------------------------------------------
__global__ void zero_pool_kernel(float* __restrict__ pooled, float* __restrict__ counts) {
    int i = blockIdx.x * blockDim.x + threadIdx.x;
    if (i < N_GRAPHS * 32) pooled[i] = 0.0f;
    if (i < N_GRAPHS)      counts[i] = 0.0f;
}

// ---------------------------------------------------------------------------
// Mean-pool accumulation: LDS bins (256 graphs x 32 feats = 32KB) per block,
// ds_add_f32 atomics, then merge to global. relu(agg2 + b2) fused on read.
// ---------------------------------------------------------------------------
__global__ void pool_kernel(const float* __restrict__ agg2,
                            const float* __restrict__ b2,
                            const int* __restrict__ batch,
                            float* __restrict__ pooled,
                            float* __restrict__ counts) {
    __shared__ float sp[N_GRAPHS * 32];
    __shared__ float sc[N_GRAPHS];
    for (int i = threadIdx.x; i < N_GRAPHS * 32; i += 256) sp[i] = 0.0f;
    for (int i = threadIdx.x; i < N_GRAPHS; i += 256)      sc[i] = 0.0f;
    __syncthreads();

    int f   = threadIdx.x & 31;      // feature
    int grp = threadIdx.x >> 5;      // 8 node-groups per block
    int per = (N_NODES + gridDim.x - 1) / gridDim.x;
    int beg = blockIdx.x * per;
    int end = beg + per;
    if (end > N_NODES) end = N_NODES;

    for (int n = beg + grp; n < end; n += 8) {
        int g = batch[n];
        float v = fmaxf(agg2[n * 32 + f] + b2[f], 0.0f);
        atomicAdd(&sp[g * 32 + f], v);
        if (f == 0) atomicAdd(&sc[g], 1.0f);
    }
    __syncthreads();

    for (int i = threadIdx.x; i < N_GRAPHS * 32; i += 256)
        if (sp[i] != 0.0f) atomicAdd(&pooled[i], sp[i]);
    for (int i = threadIdx.x; i < N_GRAPHS; i += 256)
        if (sc[i] != 0.0f) atomicAdd(&counts[i], sc[i]);
}

// ---------------------------------------------------------------------------
// Head: mean, FC (32 -> 2), log_softmax. One thread per graph.
// ---------------------------------------------------------------------------
__global__ void head_kernel(const float* __restrict__ pooled,
                            const float* __restrict__ counts,
                            const float* __restrict__ Wfc,
                            const float* __restrict__ bfc,
                            float* __restrict__ out) {
    int g = blockIdx.x * blockDim.x + threadIdx.x;
    if (g >= N_GRAPHS) return;
    float inv = 1.0f / fmaxf(counts[g], 1.0f);
    float l0 = bfc[0], l1 = bfc[1];
#pragma unroll
    for (int j = 0; j < 32; ++j) {
        float p = pooled[g * 32 + j] * inv;
        l0 += p * Wfc[j * 2 + 0];
        l1 += p * Wfc[j * 2 + 1];
    }
    float m  = fmaxf(l0, l1);
    float ls = m + logf(expf(l0 - m) + expf(l1 - m));
    out[g * 2 + 0] = l0 - ls;
    out[g * 2 + 1] = l1 - ls;
}

// ---------------------------------------------------------------------------
// Launch
// ---------------------------------------------------------------------------
extern "C" void kernel_launch(void* const* d_in, const int* in_sizes, int n_in,
                              void* d_out, int out_size, void* d_ws, size_t ws_size,
                              hipStream_t stream) {
    const float* x   = (const float*)d_in[0];
    const int*   ei  = (const int*)  d_in[1];   // [2, E] flat, int32
    const int*   bat = (const int*)  d_in[2];   // [N], int32
    const float* W1  = (const float*)d_in[3];
    const float* b1  = (const float*)d_in[4];
    const float* W2  = (const float*)d_in[5];
    const float* b2  = (const float*)d_in[6];
    const float* Wfc = (const float*)d_in[7];
    const float* bfc = (const float*)d_in[8];
    float* out = (float*)d_out;

    // Workspace carve-up (floats)
    float* ws     = (float*)d_ws;
    float* deg    = ws;                                   // N
    float* dinv   = deg  + N_NODES;                       // N
    float* h1     = dinv + N_NODES;                       // 16N
    float* agg1   = h1   + 16 * (size_t)N_NODES;          // 16N
    float* h2     = agg1 + 16 * (size_t)N_NODES;          // 32N
    float* agg2   = h2   + 32 * (size_t)N_NODES;          // 32N
    float* pooled = agg2 + 32 * (size_t)N_NODES;          // 256*32
    float* counts = pooled + N_GRAPHS * 32;               // 256

    const int T = 256;

    // Degrees and dinv
    init_deg_kernel<<<(N_NODES + T - 1) / T, T, 0, stream>>>(deg);
    count_deg_kernel<<<(N_EDGES + T - 1) / T, T, 0, stream>>>(ei, deg);
    dinv_kernel<<<(N_NODES + T - 1) / T, T, 0, stream>>>(deg, dinv);

    // Layer 1
    transform1_kernel<<<(NTILES + 7) / 8, T, 0, stream>>>(x, W1, dinv, h1, agg1);
    scatter1_kernel<<<(N_EDGES * 4 + T - 1) / T, T, 0, stream>>>(ei, dinv, h1, agg1);

    // Layer 2
    transform2_kernel<<<(NTILES + 7) / 8, T, 0, stream>>>(agg1, b1, W2, dinv, h2, agg2);
    scatter2_kernel<<<(N_EDGES * 8 + T - 1) / T, T, 0, stream>>>(ei, dinv, h2, agg2);

    // Pooling + head
    zero_pool_kernel<<<(N_GRAPHS * 32 + T - 1) / T, T, 0, stream>>>(pooled, counts);
    pool_kernel<<<100, T, 0, stream>>>(agg2, b2, bat, pooled, counts);
    head_kernel<<<1, N_GRAPHS, 0, stream>>>(pooled, counts, Wfc, bfc, out);
}